// VectorQuantization_51634096832639
// MI455X (gfx1250) — compile-verified
//
#include <hip/hip_runtime.h>
#include <hip/hip_bf16.h>

typedef float v2f __attribute__((ext_vector_type(2)));
typedef float v8f __attribute__((ext_vector_type(8)));

#define DIM 256
#define LDS_STRIDE 260            // 256 -> 260 float padding: conflict-free b64 fragment reads
#define XS_FLOATS  (16 * LDS_STRIDE)
#define BS_FLOATS  (64 * LDS_STRIDE)
#define XS_BYTES   (XS_FLOATS * 4)
#define BS_BYTES   (BS_FLOATS * 4)

// ---------------------------------------------------------------------------
// Kernel 1: c2[k] = ||vectors[k]||^2   (one wave per code; 8 floats per lane)
// ---------------------------------------------------------------------------
__global__ __launch_bounds__(256) void vq_c2_kernel(const float* __restrict__ vecs,
                                                    float* __restrict__ c2,
                                                    int K) {
    const int wave = (blockIdx.x * blockDim.x + threadIdx.x) >> 5;
    const int lane = threadIdx.x & 31;
    if (wave >= K) return;
    const float4* r4 = (const float4*)(vecs + (size_t)wave * DIM + lane * 8);
    float4 a = r4[0];
    float4 b = r4[1];
    float s = a.x * a.x + a.y * a.y + a.z * a.z + a.w * a.w
            + b.x * b.x + b.y * b.y + b.z * b.z + b.w * b.w;
#pragma unroll
    for (int m = 16; m >= 1; m >>= 1)
        s += __shfl_xor(s, m, 32);
    if (lane == 0) c2[wave] = s;
}

// ---------------------------------------------------------------------------
// CDNA5 async copy: global -> LDS, 16 bytes/lane, tracked by ASYNCcnt.
// GVS addressing: mem = SADDR(64) + VADDR(32) ; LDS dest = value in VDST VGPR.
// ---------------------------------------------------------------------------
__device__ __forceinline__ void async_g2l_b128(unsigned lds_byte_addr,
                                               unsigned g_byte_off,
                                               const float* base) {
    asm volatile("global_load_async_to_lds_b128 %0, %1, %2"
                 :: "v"(lds_byte_addr), "v"(g_byte_off), "s"(base)
                 : "memory");
}
__device__ __forceinline__ void wait_asynccnt0() {
    asm volatile("s_wait_asynccnt 0x0" ::: "memory");
}

// ---------------------------------------------------------------------------
// Kernel 2: per-row argmin over K of (||c||^2 - 2 x.c) via fp32 WMMA.
// Block: 128 threads = 4 waves; block owns 16 rows of x (A panel held in
// registers). Double-buffered 64-column codebook tiles staged with async
// global->LDS loads; wave w computes columns [t+16w, t+16w+16).
// ---------------------------------------------------------------------------
__global__ __launch_bounds__(128) void vq_argmin_kernel(const float* __restrict__ x,
                                                        const float* __restrict__ vecs,
                                                        const float* __restrict__ c2,
                                                        int* __restrict__ out,
                                                        int K) {
    extern __shared__ float smem[];
    float* xs    = smem;                                  // [16][LDS_STRIDE]
    float* rvals = smem + XS_FLOATS + 2 * BS_FLOATS;      // [4][16]
    int*   ridx  = (int*)(rvals + 64);                    // [4][16]

    const int tid     = threadIdx.x;
    const int lane    = tid & 31;
    const int wave    = tid >> 5;
    const int rowBase = blockIdx.x * 16;

    // Stage the 16x256 x-tile into LDS (one-time).
#pragma unroll
    for (int j = tid; j < 16 * 64; j += 128) {
        const int r = j >> 6, c = j & 63;
        float4 v = ((const float4*)(x + (size_t)(rowBase + r) * DIM))[c];
        float* dst = xs + r * LDS_STRIDE + c * 4;
        dst[0] = v.x; dst[1] = v.y; dst[2] = v.z; dst[3] = v.w;
    }

    // Prefetch codebook tile 0 into LDS buffer 0 (async, ASYNCcnt-tracked).
#pragma unroll
    for (int j = tid; j < 64 * 64; j += 128) {
        const int r = j >> 6, c = j & 63;
        async_g2l_b128(XS_BYTES + (unsigned)((r * LDS_STRIDE + c * 4) * 4),
                       (unsigned)((r * DIM + c * 4) * 4), vecs);
    }

    __syncthreads();   // xs visible to all waves

    // WMMA fragment addressing (ISA 7.12.2, 32-bit A 16x4 / B 4x16):
    //   lane 0-15  -> M/N = lane,    K pair {0,1}
    //   lane 16-31 -> M/N = lane-16, K pair {2,3}
    const int half = lane >> 4;
    const int mlo  = lane & 15;
    const int kk   = half * 2;

    // Hoist the wave's A panel into registers: 64 K-steps x v2f = 128 VGPRs.
    v2f afrag[64];
    {
        const float* aPtr = xs + mlo * LDS_STRIDE + kk;
#pragma unroll
        for (int i = 0; i < 64; ++i)
            afrag[i] = *(const v2f*)(aPtr + i * 4);
    }

    float bestVal[8];
    int   bestIdx[8];
#pragma unroll
    for (int i = 0; i < 8; ++i) { bestVal[i] = 3.4e38f; bestIdx[i] = 0; }

    const int nTiles = K / 64;
    for (int ti = 0; ti < nTiles; ++ti) {
        const int cur = ti & 1;
        wait_asynccnt0();      // this wave's async stores to LDS landed
        __syncthreads();       // (incl. dscnt drain) all waves' tile data visible

        // Prefetch next tile into the other buffer while we compute.
        if (ti + 1 < nTiles) {
            const unsigned gBase = (unsigned)((ti + 1) * 64 * DIM * 4);
            const unsigned lBase = XS_BYTES + (unsigned)(1 - cur) * BS_BYTES;
#pragma unroll
            for (int j = tid; j < 64 * 64; j += 128) {
                const int r = j >> 6, c = j & 63;
                async_g2l_b128(lBase + (unsigned)((r * LDS_STRIDE + c * 4) * 4),
                               gBase + (unsigned)((r * DIM + c * 4) * 4), vecs);
            }
        }

        const int   t       = ti * 64;
        const int   colBase = t + wave * 16;
        const float cn2     = c2[colBase + mlo];
        const float* bPtr   = smem + XS_FLOATS + cur * BS_FLOATS
                            + (wave * 16 + mlo) * LDS_STRIDE + kk;

        // Two independent WMMA chains (even / odd K-steps) for ILP.
        v8f acc0 = {}, acc1 = {};
#pragma unroll
        for (int i = 0; i < 64; i += 2) {
            v2f b0 = *(const v2f*)(bPtr + i * 4);
            v2f b1 = *(const v2f*)(bPtr + i * 4 + 4);
            acc0 = __builtin_amdgcn_wmma_f32_16x16x4_f32(
                false, afrag[i],     false, b0, (short)0, acc0, false, false);
            acc1 = __builtin_amdgcn_wmma_f32_16x16x4_f32(
                false, afrag[i + 1], false, b1, (short)0, acc1, false, false);
        }
        v8f acc = acc0 + acc1;

        const int nIdx = colBase + mlo;
#pragma unroll
        for (int v = 0; v < 8; ++v) {
            const float score = cn2 - 2.0f * acc[v];
            if (score < bestVal[v]) { bestVal[v] = score; bestIdx[v] = nIdx; }
        }
    }

    // Reduce across the 16 N-lanes within each half-wave (M group).
#pragma unroll
    for (int v = 0; v < 8; ++v) {
#pragma unroll
        for (int mask = 1; mask < 16; mask <<= 1) {
            const float ov = __shfl_xor(bestVal[v], mask, 32);
            const int   oi = __shfl_xor(bestIdx[v], mask, 32);
            if (ov < bestVal[v] || (ov == bestVal[v] && oi < bestIdx[v])) {
                bestVal[v] = ov; bestIdx[v] = oi;
            }
        }
    }

    __syncthreads();
    if (mlo == 0) {
#pragma unroll
        for (int v = 0; v < 8; ++v) {
            const int m = v + half * 8;
            rvals[wave * 16 + m] = bestVal[v];
            ridx [wave * 16 + m] = bestIdx[v];
        }
    }
    __syncthreads();

    // Cross-wave reduction + final store (16 rows per block).
    if (tid < 16) {
        float bv = rvals[tid];
        int   bi = ridx[tid];
#pragma unroll
        for (int w = 1; w < 4; ++w) {
            const float ov = rvals[w * 16 + tid];
            const int   oi = ridx[w * 16 + tid];
            if (ov < bv || (ov == bv && oi < bi)) { bv = ov; bi = oi; }
        }
        out[rowBase + tid] = bi;
    }
}

// ---------------------------------------------------------------------------
extern "C" void kernel_launch(void* const* d_in, const int* in_sizes, int n_in,
                              void* d_out, int out_size, void* d_ws, size_t ws_size,
                              hipStream_t stream) {
    const float* x    = (const float*)d_in[0];
    const float* vecs = (const float*)d_in[1];
    const int N = in_sizes[0] / DIM;   // 32768
    const int K = in_sizes[1] / DIM;   // 8192

    float* c2 = (float*)d_ws;          // K floats of scratch

    vq_c2_kernel<<<(K + 7) / 8, 256, 0, stream>>>(vecs, c2, K);

    const size_t ldsBytes = (size_t)(XS_FLOATS + 2 * BS_FLOATS) * sizeof(float)
                          + 64 * sizeof(float) + 64 * sizeof(int);
    vq_argmin_kernel<<<N / 16, 128, ldsBytes, stream>>>(x, vecs, c2, (int*)d_out, K);
}